// UltraMemv5Layer_21406117003388
// MI455X (gfx1250) — compile-verified
//
#include <hip/hip_runtime.h>
#include <hip/hip_bf16.h>
#include <math.h>

// ---------------- constants (match reference) ----------------
#define HD 2048
#define NN 32768
#define DK 32
#define BSZ 4096
#define NEG_INF (-3.402823466e38f)

typedef float v2f __attribute__((ext_vector_type(2)));
typedef float v8f __attribute__((ext_vector_type(8)));

// D(16x16 f32) = A(16x4 f32) x B(4x16 f32) + C   -> v_wmma_f32_16x16x4_f32
__device__ __forceinline__ v8f wmma_f32_k4(v2f a, v2f b, v8f c) {
  return __builtin_amdgcn_wmma_f32_16x16x4_f32(false, a, false, b, (short)0, c,
                                               false, false);
}

__device__ __forceinline__ float warp_sum32(float v) {
#pragma unroll
  for (int off = 16; off > 0; off >>= 1) v += __shfl_xor(v, off, 32);
  return v;
}

// register-resident sorted (descending) top-16 insert; all indices static.
__device__ __forceinline__ void insert16(float (&tv)[16], int (&ti)[16],
                                         float v, int idx) {
  if (v > tv[15]) {
    tv[15] = v; ti[15] = idx;
#pragma unroll
    for (int j = 15; j > 0; --j) {
      if (tv[j] > tv[j - 1]) {
        float tf = tv[j]; tv[j] = tv[j - 1]; tv[j - 1] = tf;
        int tq = ti[j]; ti[j] = ti[j - 1]; ti[j - 1] = tq;
      }
    }
  }
}

__device__ __forceinline__ void insert8(float (&tv)[8], int (&ti)[8], float v,
                                        int idx) {
  if (v > tv[7]) {
    tv[7] = v; ti[7] = idx;
#pragma unroll
    for (int j = 7; j > 0; --j) {
      if (tv[j] > tv[j - 1]) {
        float tf = tv[j]; tv[j] = tv[j - 1]; tv[j - 1] = tf;
        int tq = ti[j]; ti[j] = ti[j - 1]; ti[j - 1] = tq;
      }
    }
  }
}

// ------------------------------------------------------------------
// Kernel 1: Q[b][0:32]=qrow, [32:64]=qcol, [64:96]=u   (M=4096,N=96,K=2048)
// block = 64 threads (2 waves); each wave owns a 16-row batch tile, 6 col-tiles
// ------------------------------------------------------------------
__global__ __launch_bounds__(64) void k1_gemm_qu(
    const float* __restrict__ x, const float* __restrict__ q_w,
    const float* __restrict__ x_to_U_w, float* __restrict__ Q) {
  const int lane = threadIdx.x & 31;
  const int wave = threadIdx.x >> 5;
  const int b0 = (blockIdx.x * 2 + wave) * 16;
  const int m = lane & 15;
  const int kb = (lane >> 4) * 2;

  v8f acc[6];
#pragma unroll
  for (int t = 0; t < 6; ++t) acc[t] = (v8f){0, 0, 0, 0, 0, 0, 0, 0};

  const float* Wb[6];
#pragma unroll
  for (int t = 0; t < 6; ++t)
    Wb[t] = (t < 4 ? q_w + (size_t)(t * 16 + m) * HD
                   : x_to_U_w + (size_t)((t - 4) * 16 + m) * HD) + kb;
  const float* xrow = x + (size_t)(b0 + m) * HD + kb;

  for (int k0 = 0; k0 < HD; k0 += 4) {
    v2f a = *(const v2f*)(xrow + k0);
#pragma unroll
    for (int t = 0; t < 6; ++t) {
      v2f bf = *(const v2f*)(Wb[t] + k0);
      acc[t] = wmma_f32_k4(a, bf, acc[t]);
    }
  }
#pragma unroll
  for (int t = 0; t < 6; ++t) {
#pragma unroll
    for (int v = 0; v < 8; ++v) {
      int mm = v + 8 * (lane >> 4);
      Q[(size_t)(b0 + mm) * 96 + t * 16 + (lane & 15)] = acc[t][v];
    }
  }
}

// ------------------------------------------------------------------
// Kernel 2: fused score GEMM (WMMA f32) + streaming top-16 per batch row.
// grid.x = 256 batch tiles, grid.y = 2 (0: row path, 1: col path)
// block = 256 threads (8 waves); chunk = 256 columns per iteration.
// ------------------------------------------------------------------
__global__ __launch_bounds__(256) void k2_score_topk(
    const float* __restrict__ Q, const float* __restrict__ K_row,
    const float* __restrict__ K_col, const float* __restrict__ row_mix,
    const float* __restrict__ col_mix, int* __restrict__ row_idx,
    int* __restrict__ col_idx) {
  __shared__ float sc[16 * 256];  // 16 KB score staging
  __shared__ float mv[4096];      // merge values (16 rows x 16 slots x 16)
  __shared__ int mi[4096];        // merge indices

  const int tid = threadIdx.x;
  const int lane = tid & 31;
  const int wave = tid >> 5;
  const int b0 = blockIdx.x * 16;
  const int which = blockIdx.y;
  const float* Kmat = which ? K_col : K_row;
  const float mix = which ? col_mix[0] : row_mix[0];
  const int qoff = which ? 32 : 0;
  const int m = lane & 15;
  const int kb = (lane >> 4) * 2;

  // Preload A fragments (q tile 16x32 -> 8 K-steps of 4)
  v2f aF[8];
#pragma unroll
  for (int ks = 0; ks < 8; ++ks)
    aF[ks] = *(const v2f*)(Q + (size_t)(b0 + m) * 96 + qoff + ks * 4 + kb);

  float tv[16]; int ti[16];
#pragma unroll
  for (int j = 0; j < 16; ++j) { tv[j] = NEG_INF; ti[j] = 0; }
  const int r = tid >> 4, s = tid & 15;

  for (int it = 0; it < NN / 256; ++it) {
    const int nBase = it * 256;
    const float* kp0 = Kmat + (size_t)(nBase + wave * 32 + m) * DK + kb;
    const float* kp1 = kp0 + 16 * DK;
    if (it < NN / 256 - 1)
      __builtin_prefetch((const void*)(kp0 + 256 * DK), 0, 1);
    v8f a0 = (v8f){0, 0, 0, 0, 0, 0, 0, 0};
    v8f a1 = (v8f){0, 0, 0, 0, 0, 0, 0, 0};
#pragma unroll
    for (int ks = 0; ks < 8; ++ks) {
      a0 = wmma_f32_k4(aF[ks], *(const v2f*)(kp0 + ks * 4), a0);
      a1 = wmma_f32_k4(aF[ks], *(const v2f*)(kp1 + ks * 4), a1);
    }
    __syncthreads();  // previous chunk fully consumed
#pragma unroll
    for (int v = 0; v < 8; ++v) {
      int mm = v + 8 * (lane >> 4);
      sc[mm * 256 + wave * 32 + m] = a0[v] * mix;
      sc[mm * 256 + wave * 32 + 16 + m] = a1[v] * mix;
    }
    __syncthreads();
#pragma unroll
    for (int i = 0; i < 16; ++i) {
      int c = s + 16 * i;
      insert16(tv, ti, sc[r * 256 + c], nBase + c);
    }
  }
  __syncthreads();
  {
    int base = tid * 16;  // == r*256 + s*16
#pragma unroll
    for (int j = 0; j < 16; ++j) { mv[base + j] = tv[j]; mi[base + j] = ti[j]; }
  }
  __syncthreads();
  if (tid < 16) {  // one thread per batch row: merge 256 candidates -> 16
#pragma unroll
    for (int j = 0; j < 16; ++j) { tv[j] = NEG_INF; ti[j] = 0; }
    int base = tid * 256;
    for (int e = 0; e < 256; ++e) insert16(tv, ti, mv[base + e], mi[base + e]);
    int* outp = (which ? col_idx : row_idx) + (size_t)(b0 + tid) * 16;
#pragma unroll
    for (int j = 0; j < 16; ++j) outp[j] = ti[j];
  }
}

// ------------------------------------------------------------------
// Kernel M1: M1[32][8] = Bmat @ Vproj^T (K=2048). One block, 256 threads.
// ------------------------------------------------------------------
__global__ __launch_bounds__(256) void k_m1(const float* __restrict__ Bmat,
                                            const float* __restrict__ Vproj,
                                            float* __restrict__ M1) {
  const int t = threadIdx.x;
  const int rr = t >> 3, p = t & 7;
  const float* br = Bmat + (size_t)rr * HD;
  const float* vp = Vproj + (size_t)p * HD;
  float acc = 0.f;
  for (int h = 0; h < HD; ++h) acc += br[h] * vp[h];
  M1[rr * 8 + p] = acc;
}

// ------------------------------------------------------------------
// Kernel 3: per-batch finalize (one wave per batch element).
// A_sel/B_sel -> top-8 of 16x16 grid -> softmax -> emb gathers ->
// S/T rows (+top-4 sparsify) -> pv -> normalized s_acc (4096x32).
// ------------------------------------------------------------------
__global__ __launch_bounds__(256) void k3_finalize(
    const float* __restrict__ Q, const float* __restrict__ K_row,
    const float* __restrict__ K_col, const float* __restrict__ core,
    const int* __restrict__ row_idx, const int* __restrict__ col_idx,
    const float* __restrict__ row_emb, const float* __restrict__ col_emb,
    const float* __restrict__ wSr_g, const float* __restrict__ wSc_g,
    const float* __restrict__ wTr_g, const float* __restrict__ wTc_g,
    float* __restrict__ s_out) {
  __shared__ float wSr[1024], wSc[1024], wTr[1024], wTc[1024];
  __shared__ float sAB[8 * 64];  // per wave: [0:16]=A_sel, [32:48]=B_sel

  const int tid = threadIdx.x;
  const int lane = tid & 31;
  const int wave = tid >> 5;
  for (int i = tid; i < 1024; i += 256) {
    wSr[i] = wSr_g[i]; wSc[i] = wSc_g[i];
    wTr[i] = wTr_g[i]; wTc[i] = wTc_g[i];
  }
  __syncthreads();

  const int b = blockIdx.x * 8 + wave;
  const float qr = Q[(size_t)b * 96 + lane];
  const float qc = Q[(size_t)b * 96 + 32 + lane];
  const float uu = Q[(size_t)b * 96 + 64 + lane];
  const float c00 = core[0];

  for (int p = 0; p < 16; ++p) {
    int ri = row_idx[b * 16 + p];
    float av = warp_sum32(K_row[(size_t)ri * DK + lane] * qr) * c00;
    int ci = col_idx[b * 16 + p];
    float bv = warp_sum32(K_col[(size_t)ci * DK + lane] * qc);
    if (lane == p) { sAB[wave * 64 + p] = av; sAB[wave * 64 + 32 + p] = bv; }
  }
  __syncthreads();

  const float myA = sAB[wave * 64 + (lane & 15)];
  const int qb = (lane >> 4) * 8;
  float prv[8]; int pri[8];
#pragma unroll
  for (int j = 0; j < 8; ++j) {
    float bb = sAB[wave * 64 + 32 + qb + j];
    prv[j] = myA * bb;
    pri[j] = (lane & 15) * 16 + (qb + j);
  }
  // top-8 of 256 products; all lanes maintain an identical list via shuffles
  float t8v[8]; int t8i[8];
#pragma unroll
  for (int j = 0; j < 8; ++j) { t8v[j] = NEG_INF; t8i[j] = 0; }
#pragma unroll
  for (int j = 0; j < 8; ++j) {
    for (int src = 0; src < 32; ++src) {
      float v = __shfl(prv[j], src, 32);
      int pi = __shfl(pri[j], src, 32);
      insert8(t8v, t8i, v, pi);
    }
  }
  // softmax weights (t8v sorted descending -> t8v[0] is max)
  float w8[8]; float wsum = 0.f;
#pragma unroll
  for (int j = 0; j < 8; ++j) { w8[j] = expf(t8v[j] - t8v[0]); wsum += w8[j]; }
#pragma unroll
  for (int j = 0; j < 8; ++j) w8[j] /= wsum;

  // embedding gathers: lane = feature index
  float rv8[8], cv8[8];
#pragma unroll
  for (int mI = 0; mI < 8; ++mI) {
    int gr = row_idx[b * 16 + (t8i[mI] >> 4)];
    int gc = col_idx[b * 16 + (t8i[mI] & 15)];
    rv8[mI] = row_emb[(size_t)gr * 32 + lane];
    cv8[mI] = col_emb[(size_t)gc * 32 + lane];
  }
  // S_rows / T_rows: lane j computes output feature j
  float Sr[8], Tr[8];
#pragma unroll
  for (int mI = 0; mI < 8; ++mI) {
    float sv = 0.f, tvv = 0.f;
    for (int k = 0; k < 32; ++k) {
      float rv = __shfl(rv8[mI], k, 32);
      float cv = __shfl(cv8[mI], k, 32);
      sv += rv * wSr[lane * 32 + k] + cv * wSc[lane * 32 + k];
      tvv += rv * wTr[lane * 32 + k] + cv * wTc[lane * 32 + k];
    }
    Sr[mI] = sv; Tr[mI] = tvv;
  }
  // top-4-by-|v| sparsify: keep iff strict-greater-count <= 3  (== |v|>=thresh)
#pragma unroll
  for (int mI = 0; mI < 8; ++mI) {
    float aS = fabsf(Sr[mI]); int cS = 0;
    float aT = fabsf(Tr[mI]); int cT = 0;
    for (int src = 0; src < 32; ++src) {
      cS += (__shfl(aS, src, 32) > aS) ? 1 : 0;
      cT += (__shfl(aT, src, 32) > aT) ? 1 : 0;
    }
    if (cS > 3) Sr[mI] = 0.f;
    if (cT > 3) Tr[mI] = 0.f;
  }
  // pv, a, s_acc, normalize
  float sacc = 0.f;
#pragma unroll
  for (int mI = 0; mI < 8; ++mI) {
    float pv = warp_sum32(Tr[mI] * uu);
    sacc += (w8[mI] * pv) * Sr[mI];
  }
  float nrm = sqrtf(warp_sum32(sacc * sacc));
  sacc /= fmaxf(nrm, 1e-12f);
  s_out[(size_t)b * 32 + lane] = sacc;
}

// ------------------------------------------------------------------
// Kernel 4: out = G + tanh(gamma) * (G @ Vproj^T) @ Uproj^T,  G = s_acc @ Bmat
// Uses precomputed M1 = Bmat@Vproj^T so t = s_acc@M1 (rank-8).
// grid (32, 256): block covers 16 batch rows x 64 output cols.
// ------------------------------------------------------------------
__global__ __launch_bounds__(256) void k4_out(
    const float* __restrict__ sacc, const float* __restrict__ M1,
    const float* __restrict__ Bmat, const float* __restrict__ Uproj,
    const float* __restrict__ gamma, float* __restrict__ out) {
  __shared__ float sA[512];  // 16 x 32 s_acc tile
  __shared__ float sM[256];  // M1 32x8
  __shared__ float sT[128];  // t 16x8
  const int tid = threadIdx.x;
  const int b0 = blockIdx.y * 16;
  sM[tid] = M1[tid];
  sA[tid] = sacc[(size_t)b0 * 32 + tid];
  sA[256 + tid] = sacc[(size_t)b0 * 32 + 256 + tid];
  __syncthreads();
  if (tid < 128) {
    int bb = tid >> 3, p = tid & 7;
    float a = 0.f;
#pragma unroll
    for (int rr = 0; rr < 32; ++rr) a += sA[bb * 32 + rr] * sM[rr * 8 + p];
    sT[tid] = a;
  }
  __syncthreads();
  const float tg = tanhf(gamma[0]);
  const int ty = tid >> 4, tx = tid & 15;
  const int h0 = blockIdx.x * 64 + tx * 4;
  float a0 = 0, a1 = 0, a2 = 0, a3 = 0;
#pragma unroll
  for (int rr = 0; rr < 32; ++rr) {
    float4 bv = *(const float4*)(Bmat + (size_t)rr * HD + h0);
    float sv = sA[ty * 32 + rr];
    a0 += sv * bv.x; a1 += sv * bv.y; a2 += sv * bv.z; a3 += sv * bv.w;
  }
  float lr[4];
#pragma unroll
  for (int hh = 0; hh < 4; ++hh) {
    const float* up = Uproj + (size_t)(h0 + hh) * 8;
    float4 ua = *(const float4*)(up);
    float4 ub = *(const float4*)(up + 4);
    lr[hh] = sT[ty * 8 + 0] * ua.x + sT[ty * 8 + 1] * ua.y +
             sT[ty * 8 + 2] * ua.z + sT[ty * 8 + 3] * ua.w +
             sT[ty * 8 + 4] * ub.x + sT[ty * 8 + 5] * ub.y +
             sT[ty * 8 + 6] * ub.z + sT[ty * 8 + 7] * ub.w;
  }
  float4 res;
  res.x = a0 + tg * lr[0]; res.y = a1 + tg * lr[1];
  res.z = a2 + tg * lr[2]; res.w = a3 + tg * lr[3];
  *(float4*)(out + (size_t)(b0 + ty) * HD + h0) = res;
}

// ------------------------------------------------------------------
extern "C" void kernel_launch(void* const* d_in, const int* in_sizes, int n_in,
                              void* d_out, int out_size, void* d_ws,
                              size_t ws_size, hipStream_t stream) {
  const float* x = (const float*)d_in[0];
  const float* q_w = (const float*)d_in[1];
  const float* K_row = (const float*)d_in[2];
  const float* K_col = (const float*)d_in[3];
  const float* core = (const float*)d_in[4];
  const float* row_mix = (const float*)d_in[5];
  const float* col_mix = (const float*)d_in[6];
  const float* row_emb = (const float*)d_in[7];
  const float* col_emb = (const float*)d_in[8];
  const float* row_to_S_w = (const float*)d_in[9];
  const float* col_to_S_w = (const float*)d_in[10];
  const float* row_to_T_w = (const float*)d_in[11];
  const float* col_to_T_w = (const float*)d_in[12];
  const float* Bmat = (const float*)d_in[13];
  const float* x_to_U_w = (const float*)d_in[14];
  const float* Vproj_w = (const float*)d_in[15];
  const float* Uproj_w = (const float*)d_in[16];
  const float* gamma = (const float*)d_in[17];
  float* out = (float*)d_out;

  float* ws = (float*)d_ws;
  float* Q = ws;                           // 4096*96      = 393216
  int* row_idx = (int*)(ws + 393216);      // 4096*16      =  65536
  int* col_idx = (int*)(ws + 458752);      // 4096*16      =  65536
  float* s_acc = ws + 524288;              // 4096*32      = 131072
  float* M1 = ws + 655360;                 // 32*8         =    256

  k1_gemm_qu<<<128, 64, 0, stream>>>(x, q_w, x_to_U_w, Q);
  k_m1<<<1, 256, 0, stream>>>(Bmat, Vproj_w, M1);
  k2_score_topk<<<dim3(256, 2), 256, 0, stream>>>(Q, K_row, K_col, row_mix,
                                                  col_mix, row_idx, col_idx);
  k3_finalize<<<512, 256, 0, stream>>>(Q, K_row, K_col, core, row_idx, col_idx,
                                       row_emb, col_emb, row_to_S_w, col_to_S_w,
                                       row_to_T_w, col_to_T_w, s_acc);
  k4_out<<<dim3(32, 256), 256, 0, stream>>>(s_acc, M1, Bmat, Uproj_w, gamma,
                                            out);
}